// SelfAttention_7060926235165
// MI455X (gfx1250) — compile-verified
//
#include <hip/hip_runtime.h>

// ---------------------------------------------------------------------------
// Self-attention with shared-wi heads on gfx1250 (MI455X), exact f32 math via
// V_WMMA_F32_16X16X4_F32 + async global->LDS staging (ASYNCcnt double buffer).
//
// Key algebraic simplification: all heads identical =>
//   out = (qp + softmax(qp kp^T / 8) vp) @ Wo_sum + bo,
//   Wo_sum[d,m] = sum_h Wo[h*64+d, m]   (64 x 1024)
// ---------------------------------------------------------------------------

typedef float v2f __attribute__((ext_vector_type(2)));
typedef float v8f __attribute__((ext_vector_type(8)));
typedef int   v4i __attribute__((ext_vector_type(4)));

#define B_  4
#define S_  2048
#define DM  1024
#define DH  64
#define NH  16
#define BS  (B_ * S_)          // 8192 total rows
#define NT  (S_ / 16)          // 128 key tiles per batch

#define AS1 __attribute__((address_space(1)))
#define AS3 __attribute__((address_space(3)))

#if __has_builtin(__builtin_amdgcn_global_load_async_to_lds_b128)
#define HAVE_ASYNC 1
#else
#define HAVE_ASYNC 0
#endif

#if HAVE_ASYNC
#if __has_builtin(__builtin_amdgcn_s_wait_asynccnt)
#define WAIT_ASYNC(n) __builtin_amdgcn_s_wait_asynccnt(n)
#else
#define WAIT_ASYNC(n) asm volatile("s_wait_asynccnt " #n ::: "memory")
#endif
#else
#define WAIT_ASYNC(n)
#endif

static __device__ __forceinline__ v8f wmma_f32(v2f a, v2f b, v8f c) {
  // D = A(16x4,f32) * B(4x16,f32) + C(16x16,f32); exact f32 WMMA on CDNA5.
  return __builtin_amdgcn_wmma_f32_16x16x4_f32(
      /*neg_a=*/false, a, /*neg_b=*/false, b,
      /*c_mod=*/(short)0, c, /*reuse_a=*/false, /*reuse_b=*/false);
}

// 16-byte global -> LDS copy (async on CDNA5, sync fallback otherwise).
static __device__ __forceinline__ void cp16(const float* g, float* l) {
#if HAVE_ASYNC
  __builtin_amdgcn_global_load_async_to_lds_b128(
      (AS1 v4i*)g, (AS3 v4i*)l, /*imm_off=*/0, /*cpol=*/0);
#else
  *(float4*)l = *(const float4*)g;
#endif
}

// Cooperatively stage one 16x64 K tile + 16x64 V tile (128 threads,
// 256 float4 chunks per tile -> 2 chunks per thread per tile).
static __device__ __forceinline__ void stage_tiles(float* kdst, float* vdst,
                                                   const float* kg,
                                                   const float* vg, int tid) {
  const int f0 = tid * 4;            // float offset of chunk 0
  const int f1 = (tid + 128) * 4;    // float offset of chunk 1
  cp16(kg + f0, kdst + f0);
  cp16(kg + f1, kdst + f1);
  cp16(vg + f0, vdst + f0);
  cp16(vg + f1, vdst + f1);
}

// ---------------------------------------------------------------------------
// Kernel 1: projections qp/kp/vp = {q,k,v} @ wi.
// grid = (BS/16, 3), block = 128 (4 waves). Block computes one 16-row tile of
// one projection; wave w owns output cols [16w, 16w+16). K loop: 256 WMMAs.
// ---------------------------------------------------------------------------
__global__ __launch_bounds__(128) void proj_kernel(
    const float* __restrict__ q, const float* __restrict__ k,
    const float* __restrict__ v, const float* __restrict__ wi,
    float* __restrict__ ws) {
  const int lane = threadIdx.x & 31;
  const int wave = threadIdx.x >> 5;
  const int lo = lane & 15;      // A: row m / B,C: col n
  const int hi = lane >> 4;      // selects K pair (A/B) or row half (C)
  const int row0 = blockIdx.x * 16;
  const int which = blockIdx.y;  // 0 = qp, 1 = kp, 2 = vp
  const float* src = (which == 0) ? q : (which == 1) ? k : v;
  float* dst = ws + (size_t)which * (size_t)BS * DH;

  const int n = 16 * wave + lo;
  const float* arow = src + (size_t)(row0 + lo) * DM + 2 * hi;

  v8f acc = {0.f, 0.f, 0.f, 0.f, 0.f, 0.f, 0.f, 0.f};
  for (int k0 = 0; k0 < DM; k0 += 4) {
    v2f a, b;
    a[0] = arow[k0];
    a[1] = arow[k0 + 1];
    b[0] = wi[(size_t)(k0 + 2 * hi) * DH + n];
    b[1] = wi[(size_t)(k0 + 2 * hi + 1) * DH + n];
    acc = wmma_f32(a, b, acc);
  }
#pragma unroll
  for (int r = 0; r < 8; ++r)
    dst[(size_t)(row0 + r + 8 * hi) * DH + n] = acc[r];
}

// ---------------------------------------------------------------------------
// Kernel 2: Wo_sum[d,m] = sum_h Wo[h*64+d, m].  (64 x 1024 elements)
// ---------------------------------------------------------------------------
__global__ void wo_reduce_kernel(const float* __restrict__ Wo,
                                 float* __restrict__ wo_sum) {
  int idx = blockIdx.x * blockDim.x + threadIdx.x;  // idx = d*DM + m
  if (idx >= DH * DM) return;
  int d = idx / DM, m = idx - d * DM;
  float s = 0.f;
#pragma unroll
  for (int h = 0; h < NH; ++h) s += Wo[(size_t)(h * DH + d) * DM + m];
  wo_sum[idx] = s;
}

// ---------------------------------------------------------------------------
// Kernel 3: flash attention. block = 128 (4 waves), each wave owns one
// 16-query tile; the block shares double-buffered async-staged K/V tiles in
// LDS (all 4 query tiles scan the same keys of one batch).
// Online softmax state lives per lane over C-layout rows; P is bounced
// through a per-wave LDS tile to re-fragment C-layout -> A-layout.
// head = qp + (P@V)/l written to workspace.
// ---------------------------------------------------------------------------
__global__ __launch_bounds__(128) void attn_kernel(
    const float* __restrict__ qp, const float* __restrict__ kp,
    const float* __restrict__ vp, float* __restrict__ head) {
  __shared__ __align__(16) float kbuf[2][16 * DH];   // 8 KB
  __shared__ __align__(16) float vbuf[2][16 * DH];   // 8 KB
  __shared__ float pbuf[4][16 * 16];                 // 4 KB (per-wave P tiles)

  const int tid = threadIdx.x;
  const int lane = tid & 31;
  const int wave = tid >> 5;
  const int lo = lane & 15;
  const int hi = lane >> 4;
  const int batch = blockIdx.x / (S_ / 64);
  const int qrow0 = blockIdx.x * 64 + wave * 16;   // flat row in [0, BS)
  const int kbase = batch * S_;
  float* Pt = pbuf[wave];

  // Preload all 16 A-fragments of the scaled query tile (16x64, K chunks of 4).
  v2f qa[16];
  {
    const float* qrow = qp + (size_t)(qrow0 + lo) * DH + 2 * hi;
#pragma unroll
    for (int ks = 0; ks < 16; ++ks) {
      qa[ks][0] = qrow[4 * ks] * 0.125f;      // 1/sqrt(64)
      qa[ks][1] = qrow[4 * ks + 1] * 0.125f;
    }
  }

  float mrow[8], lrow[8];
#pragma unroll
  for (int r = 0; r < 8; ++r) {
    mrow[r] = -__builtin_huge_valf();
    lrow[r] = 0.f;
  }
  v8f zero8 = {0.f, 0.f, 0.f, 0.f, 0.f, 0.f, 0.f, 0.f};
  v8f acc[4] = {zero8, zero8, zero8, zero8};

  // Prologue: stage key tile 0 into buffer 0.
  stage_tiles(kbuf[0], vbuf[0], kp + (size_t)kbase * DH,
              vp + (size_t)kbase * DH, tid);

  int cur = 0;
  for (int kt = 0; kt < NT; ++kt) {
    // Issue next tile into the other buffer, then wait for the current tile:
    // the 4 newest async ops per wave are the next tile; waiting to <=4
    // guarantees all older (current-tile) ops have landed (in-order retire).
    if (kt + 1 < NT) {
      const int key1 = kbase + (kt + 1) * 16;
      stage_tiles(kbuf[cur ^ 1], vbuf[cur ^ 1], kp + (size_t)key1 * DH,
                  vp + (size_t)key1 * DH, tid);
      WAIT_ASYNC(4);
    } else {
      WAIT_ASYNC(0);
    }
    __syncthreads();  // current tile visible to all waves

    const float* kt_lds = kbuf[cur];
    const float* vt_lds = vbuf[cur];

    // ---- scores: S = (qp/8) @ kp^T   (16x16, K=64) ----
    v8f sc = zero8;
    const float* krow = kt_lds + lo * DH + 2 * hi;  // B(k,n)=kp[n][k], n=lo
#pragma unroll
    for (int ks = 0; ks < 16; ++ks) {
      v2f kb;
      kb[0] = krow[4 * ks];
      kb[1] = krow[4 * ks + 1];
      sc = wmma_f32(qa[ks], kb, sc);
    }

    // ---- online softmax update ----
    float sf[8];
#pragma unroll
    for (int r = 0; r < 8; ++r) {
      float x = sc[r];
      x = fmaxf(x, __shfl_xor(x, 1, 32));
      x = fmaxf(x, __shfl_xor(x, 2, 32));
      x = fmaxf(x, __shfl_xor(x, 4, 32));
      x = fmaxf(x, __shfl_xor(x, 8, 32));   // row max within 16-lane group
      float nm = fmaxf(mrow[r], x);
      sf[r] = __expf(mrow[r] - nm);         // exp(-inf)=0 on first tile
      float p = __expf(sc[r] - nm);
      Pt[(r + 8 * hi) * 16 + lo] = p;       // C-layout -> LDS (row-major)
      float rs = p;
      rs += __shfl_xor(rs, 1, 32);
      rs += __shfl_xor(rs, 2, 32);
      rs += __shfl_xor(rs, 4, 32);
      rs += __shfl_xor(rs, 8, 32);          // row sum
      lrow[r] = lrow[r] * sf[r] + rs;
      mrow[r] = nm;
    }
#pragma unroll
    for (int t = 0; t < 4; ++t)
#pragma unroll
      for (int r = 0; r < 8; ++r) acc[t][r] *= sf[r];

    // ---- O += P @ V  (A = P from per-wave LDS; B = V tile from shared LDS)
#pragma unroll
    for (int kc = 0; kc < 4; ++kc) {
      v2f pa;
      pa[0] = Pt[lo * 16 + 4 * kc + 2 * hi];
      pa[1] = Pt[lo * 16 + 4 * kc + 2 * hi + 1];
      const float* vr0 = vt_lds + (4 * kc + 2 * hi) * DH;
      const float* vr1 = vr0 + DH;
#pragma unroll
      for (int t = 0; t < 4; ++t) {
        v2f vb;
        vb[0] = vr0[16 * t + lo];
        vb[1] = vr1[16 * t + lo];
        acc[t] = wmma_f32(pa, vb, acc[t]);
      }
    }

    __syncthreads();  // all waves done with this tile before next overwrite
    cur ^= 1;
  }

  // ---- epilogue: head = qp + O / l ----
#pragma unroll
  for (int t = 0; t < 4; ++t)
#pragma unroll
    for (int r = 0; r < 8; ++r) {
      int m = qrow0 + r + 8 * hi;
      int n = 16 * t + lo;
      head[(size_t)m * DH + n] =
          qp[(size_t)m * DH + n] + acc[t][r] / lrow[r];
    }
}

// ---------------------------------------------------------------------------
// Kernel 4: out = head @ Wo_sum + bo.
// grid = (BS/16, 2), block = 256 (8 waves). Wave owns 4 col-tiles (64 cols).
// ---------------------------------------------------------------------------
__global__ __launch_bounds__(256) void out_kernel(
    const float* __restrict__ head, const float* __restrict__ wo_sum,
    const float* __restrict__ bo, float* __restrict__ out) {
  const int lane = threadIdx.x & 31;
  const int wave = threadIdx.x >> 5;
  const int lo = lane & 15;
  const int hi = lane >> 4;
  const int row0 = blockIdx.x * 16;
  const int col0 = blockIdx.y * 512 + wave * 64;

  v2f ha[16];
  {
    const float* hrow = head + (size_t)(row0 + lo) * DH + 2 * hi;
#pragma unroll
    for (int ks = 0; ks < 16; ++ks) {
      ha[ks][0] = hrow[4 * ks];
      ha[ks][1] = hrow[4 * ks + 1];
    }
  }

  v8f zero8 = {0.f, 0.f, 0.f, 0.f, 0.f, 0.f, 0.f, 0.f};
  v8f acc[4] = {zero8, zero8, zero8, zero8};

#pragma unroll
  for (int ks = 0; ks < 16; ++ks) {
    const float* w0 = wo_sum + (size_t)(4 * ks + 2 * hi) * DM + col0;
    const float* w1 = w0 + DM;
#pragma unroll
    for (int t = 0; t < 4; ++t) {
      v2f wb;
      wb[0] = w0[16 * t + lo];
      wb[1] = w1[16 * t + lo];
      acc[t] = wmma_f32(ha[ks], wb, acc[t]);
    }
  }

#pragma unroll
  for (int t = 0; t < 4; ++t) {
    int n = col0 + 16 * t + lo;
    float bb = bo[n];
#pragma unroll
    for (int r = 0; r < 8; ++r)
      out[(size_t)(row0 + r + 8 * hi) * DM + n] = acc[t][r] + bb;
  }
}

// ---------------------------------------------------------------------------
// Launch. Input order (setup_inputs dict): v, k, q, wi, Wo, bo.
// Workspace layout (floats): qp | kp | vp | head | wo_sum  (~8.5 MB total).
// ---------------------------------------------------------------------------
extern "C" void kernel_launch(void* const* d_in, const int* in_sizes, int n_in,
                              void* d_out, int out_size, void* d_ws,
                              size_t ws_size, hipStream_t stream) {
  (void)in_sizes; (void)n_in; (void)out_size; (void)ws_size;
  const float* v  = (const float*)d_in[0];
  const float* k  = (const float*)d_in[1];
  const float* q  = (const float*)d_in[2];
  const float* wi = (const float*)d_in[3];
  const float* Wo = (const float*)d_in[4];
  const float* bo = (const float*)d_in[5];
  float* out = (float*)d_out;
  float* ws  = (float*)d_ws;

  float* qp     = ws;
  float* kp     = ws + 1 * (size_t)BS * DH;
  float* vp     = ws + 2 * (size_t)BS * DH;
  float* head   = ws + 3 * (size_t)BS * DH;
  float* wo_sum = ws + 4 * (size_t)BS * DH;

  proj_kernel<<<dim3(BS / 16, 3), 128, 0, stream>>>(q, k, v, wi, ws);
  wo_reduce_kernel<<<dim3((DH * DM + 255) / 256), 256, 0, stream>>>(Wo, wo_sum);
  attn_kernel<<<dim3(BS / 64), 128, 0, stream>>>(qp, kp, vp, head);
  out_kernel<<<dim3(BS / 16, 2), 256, 0, stream>>>(head, wo_sum, bo, out);
}